// AttentionSchemaNetwork_48610439856445
// MI455X (gfx1250) — compile-verified
//
#include <hip/hip_runtime.h>
#include <hip/hip_bf16.h>
#include <math.h>
#include <stdint.h>

// ---------------------------------------------------------------- constants
#define H_DIM 1024
#define WS_   512
#define C_    16
#define NH_   8
#define DH_   128
#define B_    16
#define NDOCS 512

typedef __attribute__((ext_vector_type(2))) float v2f;
typedef __attribute__((ext_vector_type(8))) float v8f;

// CDNA5 async global->LDS (ASYNCcnt) path, guarded so compile never breaks.
#if defined(__AMDGCN__) && __has_builtin(__builtin_amdgcn_global_load_async_to_lds_b128) && \
    __has_builtin(__builtin_amdgcn_s_wait_asynccnt)
#define HAVE_ASYNC_LDS 1
#else
#define HAVE_ASYNC_LDS 0
#endif

#if HAVE_ASYNC_LDS
typedef int v4i_g __attribute__((vector_size(16)));                 // 16B payload type
typedef __attribute__((address_space(1))) v4i_g as1_v4i;            // global
typedef __attribute__((address_space(3))) v4i_g as3_v4i;            // LDS
#endif

__device__ __forceinline__ void cp_issue_b128(const float* g, float* l, float4& reg)
{
#if HAVE_ASYNC_LDS
    // AS1 = raw 64-bit global address; AS3 = low 32 bits of generic LDS address.
    __builtin_amdgcn_global_load_async_to_lds_b128(
        (as1_v4i*)(uintptr_t)g,
        (as3_v4i*)(unsigned)(uintptr_t)l,
        0, 0);
    (void)reg;
#else
    (void)l;
    reg = *(const float4*)g;
#endif
}

__device__ __forceinline__ void cp_commit_b128(float* l, const float4& reg)
{
#if !HAVE_ASYNC_LDS
    l[0] = reg.x; l[1] = reg.y; l[2] = reg.z; l[3] = reg.w;
#else
    (void)l; (void)reg;
#endif
}

__device__ __forceinline__ void cp_wait_all()
{
#if HAVE_ASYNC_LDS
    __builtin_amdgcn_s_wait_asynccnt(0);
#endif
}

// ============================================================== generic GEMM
// D[m,n] = alpha * ( sum_k A[m,k] * B[k,n]  + bias[n] )
//   TRANSB=1 : B[k,n] = W[n*K + k]   (W is [N,K] row-major, we use W^T)
//   TRANSB=0 : B[k,n] = W[k*N + n]   (W is [K,N] row-major)
// grid = (N/128, M/16, Z) ; block = 256 (8 waves, one 16x16 tile per wave)
// Double-buffered LDS: tile i+1 is in flight (async-to-LDS or reg-staged)
// while the 8 V_WMMA_F32_16X16X4_F32 of tile i execute.
// batched offsets: off = (z/zdiv)*hi + (z%zdiv)*lo   for A, W, bias, Out
template <bool TRANSB>
__global__ __launch_bounds__(256) void gemm16(
    const float* __restrict__ A, const float* __restrict__ W,
    const float* __restrict__ bias, float* __restrict__ Out,
    int N, int K, int lda, int ldout,
    long long zAhi, long long zAlo, long long zWhi, long long zWlo,
    long long zBhi, long long zBlo, long long zOhi, long long zOlo,
    int zdiv, float alpha)
{
    constexpr int KB = 32;
    __shared__ float sA[2][16 * 33];   // 16 x 32 padded
    __shared__ float sB[2][4224];      // TRANSB: 128x33 ; else 32x132 (both 4224)

    const int tid  = threadIdx.x;
    const int lane = tid & 31;
    const int wv   = tid >> 5;
    const int half = (lane >> 4) & 1;
    const int l16  = lane & 15;
    const int z    = blockIdx.z;
    const int nblk = blockIdx.x * 128;
    const int nloc = wv * 16 + l16;
    const int ncol = nblk + nloc;

    const long long zh = z / zdiv, zl = z % zdiv;
    A   += zh * zAhi + zl * zAlo + (long long)blockIdx.y * 16 * lda;
    W   += zh * zWhi + zl * zWlo;
    Out += zh * zOhi + zl * zOlo + (long long)blockIdx.y * 16 * ldout;
    const float* Bp = bias ? (bias + zh * zBhi + zl * zBlo) : nullptr;

    v8f acc = {};
    float4 ra = {0, 0, 0, 0};
    float4 rw[4];

    auto tile_issue = [&](int k0, float* dA, float* dB) {
        if (tid < 128) {   // A tile: 16 rows x 32 k
            const int r = tid >> 3, c = (tid & 7) * 4;
            cp_issue_b128(A + (long long)r * lda + k0 + c, &dA[r * 33 + c], ra);
        }
        if (TRANSB) {      // W tile: 128 rows(n) x 32 cols(k)
            #pragma unroll
            for (int it = 0; it < 4; ++it) {
                const int r = it * 32 + (tid >> 3), c = (tid & 7) * 4;
                cp_issue_b128(W + (long long)(nblk + r) * K + k0 + c,
                              &dB[r * 33 + c], rw[it]);
            }
        } else {           // W tile: 32 rows(k) x 128 cols(n)
            #pragma unroll
            for (int it = 0; it < 4; ++it) {
                const int r = it * 8 + (tid >> 5), c = lane * 4;
                cp_issue_b128(W + (long long)(k0 + r) * N + nblk + c,
                              &dB[r * 132 + c], rw[it]);
            }
        }
    };
    auto tile_commit = [&](float* dA, float* dB) {
#if !HAVE_ASYNC_LDS
        if (tid < 128) {
            const int r = tid >> 3, c = (tid & 7) * 4;
            cp_commit_b128(&dA[r * 33 + c], ra);
        }
        if (TRANSB) {
            #pragma unroll
            for (int it = 0; it < 4; ++it) {
                const int r = it * 32 + (tid >> 3), c = (tid & 7) * 4;
                cp_commit_b128(&dB[r * 33 + c], rw[it]);
            }
        } else {
            #pragma unroll
            for (int it = 0; it < 4; ++it) {
                const int r = it * 8 + (tid >> 5), c = lane * 4;
                cp_commit_b128(&dB[r * 132 + c], rw[it]);
            }
        }
#endif
        (void)dA; (void)dB;
    };

    const int nt = K / KB;
    tile_issue(0, sA[0], sB[0]);
    cp_wait_all();
    tile_commit(sA[0], sB[0]);
    __syncthreads();

    for (int i = 0; i < nt; ++i) {
        float* cA = sA[i & 1];
        float* cB = sB[i & 1];
        float* nA = sA[(i + 1) & 1];
        float* nB = sB[(i + 1) & 1];
        const bool more = (i + 1) < nt;
        if (more) tile_issue((i + 1) * KB, nA, nB);   // prefetch next tile

        // ---- 8 x V_WMMA_F32_16X16X4_F32 from the current LDS tile
        // A 16x4 f32 layout (ISA 7.12.2): lanes 0-15 hold K=0,1 ; lanes 16-31 K=2,3
        #pragma unroll
        for (int kk = 0; kk < KB; kk += 4) {
            v2f a, b;
            a.x = cA[l16 * 33 + kk + half * 2 + 0];
            a.y = cA[l16 * 33 + kk + half * 2 + 1];
            if (TRANSB) {
                b.x = cB[nloc * 33 + kk + half * 2 + 0];
                b.y = cB[nloc * 33 + kk + half * 2 + 1];
            } else {
                b.x = cB[(kk + half * 2 + 0) * 132 + nloc];
                b.y = cB[(kk + half * 2 + 1) * 132 + nloc];
            }
            acc = __builtin_amdgcn_wmma_f32_16x16x4_f32(
                false, a, false, b, (short)0, acc, false, false);
        }

        if (more) { cp_wait_all(); tile_commit(nA, nB); }
        __syncthreads();
    }

    const float bb = Bp ? Bp[ncol] : 0.0f;
    #pragma unroll
    for (int v = 0; v < 8; ++v) {
        const int m = v + half * 8;      // C/D layout: VGPR v -> M=v (lanes<16), v+8
        Out[(long long)m * ldout + ncol] = alpha * (acc[v] + bb);
    }
}

// =================================================== LayerNorm(512) + ReLU
__global__ __launch_bounds__(256) void ln_relu_k(
    const float* __restrict__ xin, float* __restrict__ xout,
    const float* __restrict__ g, const float* __restrict__ be)
{
    __shared__ float red[256];
    const int b = blockIdx.x, t = threadIdx.x;
    const float v0 = xin[b * WS_ + t], v1 = xin[b * WS_ + 256 + t];
    red[t] = v0 + v1; __syncthreads();
    for (int s = 128; s > 0; s >>= 1) { if (t < s) red[t] += red[t + s]; __syncthreads(); }
    const float mu = red[0] / (float)WS_;
    __syncthreads();
    red[t] = (v0 - mu) * (v0 - mu) + (v1 - mu) * (v1 - mu); __syncthreads();
    for (int s = 128; s > 0; s >>= 1) { if (t < s) red[t] += red[t + s]; __syncthreads(); }
    const float rstd = rsqrtf(red[0] / (float)WS_ + 1e-5f);
    xout[b * WS_ + t]       = fmaxf((v0 - mu) * rstd * g[t] + be[t], 0.0f);
    xout[b * WS_ + 256 + t] = fmaxf((v1 - mu) * rstd * g[t + 256] + be[t + 256], 0.0f);
}

// ========================================================= softmax over 512
__global__ __launch_bounds__(256) void softmax512_k(float* __restrict__ s)
{
    __shared__ float red[256];
    float* p = s + (long long)blockIdx.x * NDOCS;
    const int t = threadIdx.x;
    const float v0 = p[t], v1 = p[t + 256];
    red[t] = fmaxf(v0, v1); __syncthreads();
    for (int st = 128; st > 0; st >>= 1) { if (t < st) red[t] = fmaxf(red[t], red[t + st]); __syncthreads(); }
    const float mx = red[0];
    __syncthreads();
    const float e0 = expf(v0 - mx), e1 = expf(v1 - mx);
    red[t] = e0 + e1; __syncthreads();
    for (int st = 128; st > 0; st >>= 1) { if (t < st) red[t] += red[t + st]; __syncthreads(); }
    const float inv = 1.0f / red[0];
    p[t] = e0 * inv; p[t + 256] = e1 * inv;
}

// ==================================== split-K reduce: sum 4 parts + bias + act
template <int ACT>  // 0 none, 1 relu, 2 tanh
__global__ __launch_bounds__(256) void reduce4_k(
    const float* __restrict__ parts, const float* __restrict__ bias,
    float* __restrict__ out, int total, int ncol)
{
    const int i = blockIdx.x * 256 + threadIdx.x;
    if (i >= total) return;
    float v = parts[i] + parts[i + total] + parts[i + 2 * total] + parts[i + 3 * total];
    v += bias[i % ncol];
    if (ACT == 1) v = fmaxf(v, 0.0f);
    if (ACT == 2) v = tanhf(v);
    out[i] = v;
}

// ======================= final gate: logits[16,16] = t @ cg_w2^T + b, softmax
__global__ __launch_bounds__(256) void gate_cs_k(
    const float* __restrict__ tin, const float* __restrict__ w,
    const float* __restrict__ bias, float* __restrict__ cs)
{
    __shared__ float lg[256];
    const int tid = threadIdx.x, b = tid >> 4, j = tid & 15;
    const float* tr = tin + b * H_DIM;
    const float* wr = w + j * H_DIM;
    float acc = bias[j];
    for (int k = 0; k < H_DIM; k += 4)
        acc += tr[k]*wr[k] + tr[k+1]*wr[k+1] + tr[k+2]*wr[k+2] + tr[k+3]*wr[k+3];
    lg[tid] = acc; __syncthreads();
    float mx = -1e30f;
    for (int q = 0; q < 16; ++q) mx = fmaxf(mx, lg[b * 16 + q]);
    __syncthreads();
    const float e = expf(acc - mx);
    lg[tid] = e; __syncthreads();
    float sum = 0.0f;
    for (int q = 0; q < 16; ++q) sum += lg[b * 16 + q];
    cs[tid] = e / sum;
}

// ============================== pack past/present/future from sp [16,3072]
__global__ __launch_bounds__(256) void pack_ppf_k(
    const float* __restrict__ sp, float* __restrict__ out)
{
    const int i = blockIdx.x * 256 + threadIdx.x;   // 49152 total
    const int part = i >> 14, rem = i & 16383;
    const int b = rem >> 10, d = rem & 1023;
    out[i] = sp[b * 3072 + part * 1024 + d];
}

// =========================================================== host launcher
extern "C" void kernel_launch(void* const* d_in, const int* in_sizes, int n_in,
                              void* d_out, int out_size, void* d_ws, size_t ws_size,
                              hipStream_t stream)
{
    const float* ca    = (const float*)d_in[0];   // [16,1024]
    const float* docs  = (const float*)d_in[1];   // [16,512,1024]
    const float* tr_w1 = (const float*)d_in[3];   // [512,1024]
    const float* tr_b1 = (const float*)d_in[4];
    const float* ln_g  = (const float*)d_in[5];
    const float* ln_b  = (const float*)d_in[6];
    const float* tr_w2 = (const float*)d_in[7];   // [16384,512]
    const float* tr_b2 = (const float*)d_in[8];
    const float* ipw   = (const float*)d_in[9];   // [16,3072,1024]
    const float* ipb   = (const float*)d_in[10];  // [16,3072]
    const float* out_w = (const float*)d_in[11];  // [16,1024,1024]
    const float* out_b = (const float*)d_in[12];
    const float* sp_w1 = (const float*)d_in[13];  // [1024,16384]
    const float* sp_b1 = (const float*)d_in[14];
    const float* sp_w2 = (const float*)d_in[15];  // [3072,1024]
    const float* sp_b2 = (const float*)d_in[16];
    const float* cg_w1 = (const float*)d_in[17];  // [1024,3072]
    const float* cg_b1 = (const float*)d_in[18];
    const float* cg_w2 = (const float*)d_in[19];  // [16,1024]
    const float* cg_b2 = (const float*)d_in[20];
    float* out = (float*)d_out;

    // workspace layout (floats)
    float* w = (float*)d_ws;
    const long long OFF_X0 = 0, OFF_X = 8192, OFF_FEAT = 16384,
                    OFF_QH = 278528, OFF_QT = 540672, OFF_SC = 2637824,
                    OFF_SP1 = 3686400, OFF_SP = 3702784, OFF_TB = 3751936,
                    OFF_P9 = 3768320, OFF_P11 = 3833856, TOT = 3899392;
    if (ws_size < (size_t)TOT * sizeof(float)) return;
    float* x0   = w + OFF_X0;
    float* x    = w + OFF_X;
    float* feat = w + OFF_FEAT;
    float* qh   = w + OFF_QH;   // reused as o
    float* qt   = w + OFF_QT;   // reused as ctx
    float* sc   = w + OFF_SC;
    float* sp1  = w + OFF_SP1;
    float* sp   = w + OFF_SP;
    float* tb   = w + OFF_TB;
    float* p9   = w + OFF_P9;
    float* p11  = w + OFF_P11;
    float* ctrl = out + 49408;  // d_out: past|present|future|cs|controlled
    float* csO  = out + 49152;

    const float SC = 0.08838834764831845f;  // 1/sqrt(128)

    // G1: x0 = ca @ tr_w1^T + b1                  [16,512] K=1024
    gemm16<true><<<dim3(4,1,1),256,0,stream>>>(ca,tr_w1,tr_b1,x0, 512,1024,1024,512,
        0,0,0,0,0,0,0,0, 1, 1.0f);
    // LN + ReLU
    ln_relu_k<<<16,256,0,stream>>>(x0,x,ln_g,ln_b);
    // G2: feat = x @ tr_w2^T + b2                 [16,16384] K=512
    gemm16<true><<<dim3(128,1,1),256,0,stream>>>(x,tr_w2,tr_b2,feat, 16384,512,512,16384,
        0,0,0,0,0,0,0,0, 1, 1.0f);
    // G3: qh[c][b][d] = SC*(feat_c @ wq_c^T + bq_c)   per c, [16,1024] K=1024
    gemm16<true><<<dim3(8,1,16),256,0,stream>>>(feat,ipw,ipb,qh, 1024,1024,16384,1024,
        1024,0, 3145728,0, 3072,0, 16384,0, 1, SC);
    // G4: qt[c,h][b][e] = qh_ch @ wk_ch            per (c,h), [16,1024] K=128, B row-major
    gemm16<false><<<dim3(8,1,128),256,0,stream>>>(qh,ipw+1048576,nullptr,qt, 1024,128,1024,1024,
        16384,128, 3145728,131072, 0,0, 131072,16384, 8, 1.0f);
    // G5: scores[b][z][n] = qt_b @ docs_b^T        per b, M=128 N=512 K=1024
    gemm16<true><<<dim3(4,8,16),256,0,stream>>>(qt,docs,nullptr,sc, 512,1024,16384,512,
        1024,0, 524288,0, 0,0, 65536,0, 1, 1.0f);
    // (qh.bk row-constant dropped: softmax is shift-invariant per row)
    softmax512_k<<<2048,256,0,stream>>>(sc);
    // G6: ctx[b][z][e] = attn_b @ docs_b           per b, M=128 N=1024 K=512, B row-major
    gemm16<false><<<dim3(8,8,16),256,0,stream>>>(sc,docs,nullptr,qt, 1024,512,512,1024,
        65536,0, 524288,0, 0,0, 131072,0, 1, 1.0f);
    // G7: o[b][z*128+d] = ctx_ch @ wv_ch^T + bv    per (c,h), [16,128] K=1024
    gemm16<true><<<dim3(1,1,128),256,0,stream>>>(qt,ipw+2097152,ipb+2048,qh, 128,1024,131072,16384,
        8192,1024, 3145728,131072, 3072,128, 1024,128, 8, 1.0f);
    // G8: controlled = o_c @ out_w_c^T + out_b_c   per c, [16,1024] K=1024 -> d_out
    gemm16<true><<<dim3(8,1,16),256,0,stream>>>(qh,out_w,out_b,ctrl, 1024,1024,16384,16384,
        1024,0, 1048576,0, 1024,0, 1024,0, 1, 1.0f);
    // G9 (split-K x4): combined @ sp_w1^T          [16,1024] K=16384
    gemm16<true><<<dim3(8,1,4),256,0,stream>>>(ctrl,sp_w1,nullptr,p9, 1024,4096,16384,1024,
        4096,0, 4096,0, 0,0, 16384,0, 1, 1.0f);
    reduce4_k<1><<<64,256,0,stream>>>(p9,sp_b1,sp1, 16384,1024);
    // G10: sp = sp1 @ sp_w2^T + b2                 [16,3072] K=1024
    gemm16<true><<<dim3(24,1,1),256,0,stream>>>(sp1,sp_w2,sp_b2,sp, 3072,1024,1024,3072,
        0,0,0,0,0,0,0,0, 1, 1.0f);
    // G11 (split-K x4): tanh(sp @ cg_w1^T + b1)    [16,1024] K=3072
    gemm16<true><<<dim3(8,1,4),256,0,stream>>>(sp,cg_w1,nullptr,p11, 1024,768,3072,1024,
        768,0, 768,0, 0,0, 16384,0, 1, 1.0f);
    reduce4_k<2><<<64,256,0,stream>>>(p11,cg_b1,tb, 16384,1024);
    // gate: cs = softmax(tb @ cg_w2^T + b2)        [16,16]
    gate_cs_k<<<1,256,0,stream>>>(tb,cg_w2,cg_b2,csO);
    // past/present/future
    pack_ppf_k<<<192,256,0,stream>>>(sp,out);
}